// Maj3_1580547975210
// MI455X (gfx1250) — compile-verified
//
#include <hip/hip_runtime.h>

typedef __attribute__((ext_vector_type(2))) float v2f;
typedef __attribute__((ext_vector_type(8))) float v8f;

namespace {
constexpr int kC = 64, kH = 56, kW = 56, kO = 64;
constexpr int kHW = kH * kW;          // 3136
constexpr int kOStride = 3 * 3 * 64;  // 576 floats per output channel in weights
constexpr int kZeroOfs = kC * 144;    // LDS offset of the 48-float zero block
}

// One wave = one 16(w) x 16(o) output tile at fixed (b, h).
// Block = 128 threads (4 waves) covering w tiles {0,16,32,48} for one o-tile.
__global__ __launch_bounds__(128, 4) void maj3_wmma_kernel(
    const float* __restrict__ x,    // (B, C, H, W)
    const float* __restrict__ wt,   // (O, Kw, Kh, C)
    float* __restrict__ out)        // (B, O, H, W)
{
  // Weights staged as [c][kh][kw][o_rel] (conflict-free inner reads) plus a
  // 48-float zero block used as the B operand for out-of-image kh rows.
  __shared__ float wlds[kC * 144 + 48];

  const int h   = blockIdx.x;        // 0..55
  const int o0  = blockIdx.y << 4;   // 0,16,32,48
  const int b   = blockIdx.z;        // 0..3
  const int tid = threadIdx.x;

  // ---- cooperative, fully-coalesced weight staging for this o-tile ----
  {
    const float* src = wt + o0 * kOStride;
    #pragma unroll
    for (int it = 0; it < (16 * kOStride) / 128; ++it) {
      const int r    = it * 128 + tid;      // contiguous in global memory
      const int o    = r / kOStride;
      const int rem  = r - o * kOStride;    // kw*192 + kh*64 + c
      const int kw   = rem / 192;
      const int rem2 = rem - kw * 192;
      const int kh   = rem2 >> 6;
      const int c    = rem2 & 63;
      wlds[((c * 3 + kh) * 3 + kw) * 16 + o] = src[r];
    }
    if (tid < 48) wlds[kZeroOfs + tid] = 0.0f;   // zero B block
  }
  __syncthreads();

  const int lane = tid & 31;
  const int lo   = lane & 15;        // o_rel for B/D, w-row for A
  const int hi   = lane >> 4;        // lane-half: selects K pair / M-half
  const int w0   = (tid >> 5) << 4;  // wave id * 16

  // A-matrix 16x4 f32 layout: lanes0-15 v0=K0,v1=K1 ; lanes16-31 v0=K2,v1=K3.
  int   col0 = w0 - 1 + lo + (hi << 1);              // kw0 (lower) / kw2 (upper)
  float msk0 = (col0 >= 0 && col0 < kW) ? 1.0f : 0.0f;
  col0 = (col0 < 0) ? 0 : ((col0 >= kW) ? kW - 1 : col0);
  int   col1 = w0 + lo;                               // kw1 (lower half only)
  float msk1 = (hi == 0 && col1 < kW) ? 1.0f : 0.0f;  // upper half: K3 == 0
  col1 = (col1 >= kW) ? kW - 1 : col1;

  // Per-kh row validity folded into lane masks; invalid rows -> A = +0 and
  // B = LDS zero block, so every product is +0*+0 = +0 and m = +0 exactly
  // (sign bit 0, matching the reference's sign(0) = 0 contribution).
  int   rofs[3];     // clamped row * kW  (byte-safe source row offset)
  float am0[3], am1[3];
  int   nvalid = 0;
  #pragma unroll
  for (int kh = 0; kh < 3; ++kh) {
    const int  row   = h + kh - 1;
    const bool valid = (row >= 0) && (row < kH);
    const float rv   = valid ? 1.0f : 0.0f;
    rofs[kh] = (valid ? row : 0) * kW;
    am0[kh]  = msk0 * rv;
    am1[kh]  = msk1 * rv;
    nvalid  += valid ? 1 : 0;
  }

  const int bofs0 = (hi << 5) + lo;   // (hi?2:0)*16 + o_rel  (K0/K2 row of B)
  const int bofs1 = 16 + lo;          // K1 row (K3 don't-care: A's K3 is +0)

  int acc[8];
  #pragma unroll
  for (int i = 0; i < 8; ++i) acc[i] = 0;

  const float* xb = x + (size_t)b * kC * kHW;

  #pragma unroll 2
  for (int c = 0; c < kC; ++c) {
    const float* xplane = xb + c * kHW;
    if (c + 4 < kC)   // speculative prefetch of a future channel plane
      __builtin_prefetch(xplane + 4 * kHW + rofs[1] + col0, 0, 0);

    #pragma unroll
    for (int kh = 0; kh < 3; ++kh) {
      // wave-uniform scalar select of the B group (real weights vs zeros)
      const int boff = (am0[kh] != 0.0f || am1[kh] != 0.0f || nvalid == 3)
                           ? (c * 144 + kh * 48)
                           : kZeroOfs;
      const int bsel = ((h + kh - 1) >= 0 && (h + kh - 1) < kH)
                           ? (c * 144 + kh * 48)
                           : kZeroOfs;
      (void)boff;
      const float* xrow = xplane + rofs[kh];

      v2f a, bm;
      a.x  = xrow[col0] * am0[kh];   // lanes 0-15 consecutive -> coalesced
      a.y  = xrow[col1] * am1[kh];
      bm.x = wlds[bsel + bofs0];     // conflict-free DS reads
      bm.y = wlds[bsel + bofs1];

      v8f cz = {0.f, 0.f, 0.f, 0.f, 0.f, 0.f, 0.f, 0.f};
      // m[w_tile, o_tile] = A(16x4) x B(4x16), K = {kw0,kw1,kw2,0}
      v8f d = __builtin_amdgcn_wmma_f32_16x16x4_f32(
          false, a, false, bm, (short)0, cz, false, false);

      // sign-population: acc += (bits >> 31)  == -1 per negative m
      #pragma unroll
      for (int i = 0; i < 8; ++i)
        acc[i] += (__float_as_int(d[i]) >> 31);
    }
  }

  // sum sign(m) over (c, valid kh) = V + 2*acc  (V = #terms, all nonzero a.s.)
  const int vbase = kC * nvalid;

  // D layout: vgpr i, lane -> (M = i + 8*hi, N = lo) = (w = w0+8*hi+i, o = o0+lo)
  const int o      = o0 + lo;
  const int wstart = w0 + (hi << 3);
  if (wstart < kW) {             // partial tile (w0=48): upper half is w>=56
    float* dst = out + (((size_t)(b * kO + o) * kH) + h) * kW + wstart;
    float4 r0, r1;
    r0.x = (float)(vbase + 2 * acc[0]);
    r0.y = (float)(vbase + 2 * acc[1]);
    r0.z = (float)(vbase + 2 * acc[2]);
    r0.w = (float)(vbase + 2 * acc[3]);
    r1.x = (float)(vbase + 2 * acc[4]);
    r1.y = (float)(vbase + 2 * acc[5]);
    r1.z = (float)(vbase + 2 * acc[6]);
    r1.w = (float)(vbase + 2 * acc[7]);
    *reinterpret_cast<float4*>(dst)     = r0;   // 32B-aligned (rows are 56*4B)
    *reinterpret_cast<float4*>(dst + 4) = r1;
  }
}

extern "C" void kernel_launch(void* const* d_in, const int* in_sizes, int n_in,
                              void* d_out, int out_size, void* d_ws, size_t ws_size,
                              hipStream_t stream) {
  (void)in_sizes; (void)n_in; (void)d_ws; (void)ws_size; (void)out_size;
  const float* x  = (const float*)d_in[0];   // (4,64,56,56) f32
  const float* wt = (const float*)d_in[1];   // (64,3,3,64) f32
  float* out      = (float*)d_out;           // (4,64,56,56) f32

  dim3 grid(kH, kO / 16, 4);   // (h, o_tile, b) = (56, 4, 4)
  dim3 block(128);             // 4 waves of 32
  maj3_wmma_kernel<<<grid, block, 0, stream>>>(x, wt, out);
}